// MemStream_20057497272718
// MI455X (gfx1250) — compile-verified
//
#include <hip/hip_runtime.h>

#define IN_DIM   256
#define OUT_DIM  512
#define MEM_LEN  131072
#define K_SEL    16

typedef __attribute__((ext_vector_type(2))) float v2f;
typedef __attribute__((ext_vector_type(4))) float v4f;
typedef __attribute__((ext_vector_type(8))) float v8f;

// ---------------------------------------------------------------------------
// Kernel A: enc = tanh(xn @ W + b), xn = (data-mean)/std (0 where std==0).
// One block, 256 threads = 8 wave32s. Wave w owns columns [w*64, w*64+64):
// 4 N-tiles of 16 columns, 4 live f32 accumulators. Outer loop over 64
// K-steps (K=4 each); A fragment built once per K-step with branchless
// v_cndmask selects (xn loaded unconditionally via one ds_load_b128), then
// 4 independent V_WMMA_F32_16X16X4_F32 ops. Only M=0 of the 16-row A is live.
// ---------------------------------------------------------------------------
__global__ __launch_bounds__(256)
void memstream_encoder(const float* __restrict__ data,
                       const float* __restrict__ mean,
                       const float* __restrict__ stdv,
                       const float* __restrict__ W,
                       const float* __restrict__ bias,
                       float* __restrict__ enc) {
    __shared__ float xn[IN_DIM];
    const int t = threadIdx.x;
    {
        float s = stdv[t];
        xn[t] = (s == 0.0f) ? 0.0f : (data[t] - mean[t]) / s;
    }
    __syncthreads();

    const int wave = t >> 5;
    const int lane = t & 31;
    const int nsub = lane & 15;            // N within tile (both half-waves)
    const int half = lane >> 4;            // 0: K rows kb,kb+1; 1: kb+2,kb+3
    const int col0 = wave * 64 + nsub;     // tile 0 column for this lane

#if __has_builtin(__builtin_amdgcn_wmma_f32_16x16x4_f32)
    const bool rowlane = (lane < 16);

    // Bias into C row 0 (unconditional loads, branchless select).
    float bb0 = bias[col0];
    float bb1 = bias[col0 + 16];
    float bb2 = bias[col0 + 32];
    float bb3 = bias[col0 + 48];
    v8f c0 = {}, c1 = {}, c2 = {}, c3 = {};
    c0[0] = rowlane ? bb0 : 0.0f;
    c1[0] = rowlane ? bb1 : 0.0f;
    c2[0] = rowlane ? bb2 : 0.0f;
    c3[0] = rowlane ? bb3 : 0.0f;

    const v4f* __restrict__ xn4 = (const v4f*)xn;
    for (int kk = 0; kk < 64; ++kk) {
        // One uniform LDS b128 load; A built with register selects only.
        v4f x = xn4[kk];
        float a0 = (lane == 0) ? x.x : ((lane == 16) ? x.z : 0.0f);
        float a1 = (lane == 0) ? x.y : ((lane == 16) ? x.w : 0.0f);
        v2f a = {a0, a1};

        const int krow = kk * 4 + half * 2;
        const float* __restrict__ Wr0 = W + (size_t)(krow + 0) * OUT_DIM + col0;
        const float* __restrict__ Wr1 = W + (size_t)(krow + 1) * OUT_DIM + col0;
        v2f bA = {Wr0[0],  Wr1[0]};
        v2f bB = {Wr0[16], Wr1[16]};
        v2f bC = {Wr0[32], Wr1[32]};
        v2f bD = {Wr0[48], Wr1[48]};

        c0 = __builtin_amdgcn_wmma_f32_16x16x4_f32(false, a, false, bA, (short)0, c0, false, false);
        c1 = __builtin_amdgcn_wmma_f32_16x16x4_f32(false, a, false, bB, (short)0, c1, false, false);
        c2 = __builtin_amdgcn_wmma_f32_16x16x4_f32(false, a, false, bC, (short)0, c2, false, false);
        c3 = __builtin_amdgcn_wmma_f32_16x16x4_f32(false, a, false, bD, (short)0, c3, false, false);
    }

    if (lane < 16) {
        enc[col0]      = tanhf(c0[0]);
        enc[col0 + 16] = tanhf(c1[0]);
        enc[col0 + 32] = tanhf(c2[0]);
        enc[col0 + 48] = tanhf(c3[0]);
    }
#else
    // Scalar fallback: thread t computes columns t and t+256.
    for (int j = t; j < OUT_DIM; j += 256) {
        float acc = bias[j];
        for (int k = 0; k < IN_DIM; ++k) acc += xn[k] * W[k * OUT_DIM + j];
        enc[j] = tanhf(acc);
    }
#endif
}

// ---------------------------------------------------------------------------
// Kernel B: dist[row] = sum_j |memory[row][j] - enc[j]|.  Bandwidth-bound
// stream of 256 MB fp32 (~11 us at 23.3 TB/s). One row per wave32; lane holds
// its 16 fixed enc values in registers (indices are row-invariant);
// non-temporal b128 loads (256 MB > 192 MB L2, single use per pass);
// 5-step shfl_xor wave32 reduction.
// ---------------------------------------------------------------------------
__global__ __launch_bounds__(256)
void memstream_dist(const float* __restrict__ memory,
                    const float* __restrict__ enc,
                    float* __restrict__ dist) {
    const int lane = threadIdx.x & 31;
    const int gw   = (int)((blockIdx.x * blockDim.x + threadIdx.x) >> 5);
    const int nw   = (int)((gridDim.x * blockDim.x) >> 5);

    const v4f* __restrict__ enc4 = (const v4f*)enc;
    v4f e0 = enc4[0 * 32 + lane];
    v4f e1 = enc4[1 * 32 + lane];
    v4f e2 = enc4[2 * 32 + lane];
    v4f e3 = enc4[3 * 32 + lane];

    for (int row = gw; row < MEM_LEN; row += nw) {
        const v4f* __restrict__ mrow = (const v4f*)(memory + (size_t)row * OUT_DIM);
        v4f m0 = __builtin_nontemporal_load(&mrow[0 * 32 + lane]);
        v4f m1 = __builtin_nontemporal_load(&mrow[1 * 32 + lane]);
        v4f m2 = __builtin_nontemporal_load(&mrow[2 * 32 + lane]);
        v4f m3 = __builtin_nontemporal_load(&mrow[3 * 32 + lane]);

        float acc =
            fabsf(m0.x - e0.x) + fabsf(m0.y - e0.y) + fabsf(m0.z - e0.z) + fabsf(m0.w - e0.w) +
            fabsf(m1.x - e1.x) + fabsf(m1.y - e1.y) + fabsf(m1.z - e1.z) + fabsf(m1.w - e1.w) +
            fabsf(m2.x - e2.x) + fabsf(m2.y - e2.y) + fabsf(m2.z - e2.z) + fabsf(m2.w - e2.w) +
            fabsf(m3.x - e3.x) + fabsf(m3.y - e3.y) + fabsf(m3.z - e3.z) + fabsf(m3.w - e3.w);

        #pragma unroll
        for (int off = 16; off > 0; off >>= 1)
            acc += __shfl_xor(acc, off, 32);

        if (lane == 0) dist[row] = acc;
    }
}

// ---------------------------------------------------------------------------
// Kernel C: deterministic top-16 smallest (16 argmin passes, ties -> lower
// index) + weighted sum loss. One block of 1024 threads.
// ---------------------------------------------------------------------------
__global__ __launch_bounds__(1024)
void memstream_topk(const float* __restrict__ dist,
                    const float* __restrict__ exp_w,
                    float* __restrict__ out) {
    __shared__ float sVal[1024];
    __shared__ int   sIdx[1024];
    __shared__ int   found[K_SEL];
    __shared__ float sLoss;

    const int t = threadIdx.x;
    if (t == 0) sLoss = 0.0f;
    __syncthreads();

    for (int k = 0; k < K_SEL; ++k) {
        float bestV = __FLT_MAX__;
        int   bestI = MEM_LEN;  // sentinel larger than any real index
        for (int i = t; i < MEM_LEN; i += 1024) {
            bool skip = false;
            for (int f = 0; f < k; ++f) skip |= (found[f] == i);
            float v = dist[i];
            if (!skip && (v < bestV || (v == bestV && i < bestI))) {
                bestV = v; bestI = i;
            }
        }
        sVal[t] = bestV;
        sIdx[t] = bestI;
        __syncthreads();
        for (int s = 512; s > 0; s >>= 1) {
            if (t < s) {
                float ov = sVal[t + s];
                int   oi = sIdx[t + s];
                if (ov < sVal[t] || (ov == sVal[t] && oi < sIdx[t])) {
                    sVal[t] = ov; sIdx[t] = oi;
                }
            }
            __syncthreads();
        }
        if (t == 0) {
            found[k] = sIdx[0];
            sLoss += sVal[0] * exp_w[k];
        }
        __syncthreads();
    }

    if (t == 0) {
        float wsum = 0.0f;
        for (int k = 0; k < K_SEL; ++k) wsum += exp_w[k];
        out[0] = sLoss / wsum;
    }
}

// ---------------------------------------------------------------------------
extern "C" void kernel_launch(void* const* d_in, const int* in_sizes, int n_in,
                              void* d_out, int out_size, void* d_ws, size_t ws_size,
                              hipStream_t stream) {
    const float* data   = (const float*)d_in[0];
    const float* mean   = (const float*)d_in[1];
    const float* stdv   = (const float*)d_in[2];
    const float* memory = (const float*)d_in[3];
    const float* W      = (const float*)d_in[4];
    const float* bias   = (const float*)d_in[5];
    const float* exp_w  = (const float*)d_in[6];
    float*       out    = (float*)d_out;

    float* enc  = (float*)d_ws;          // [OUT_DIM]
    float* dist = enc + OUT_DIM;         // [MEM_LEN]

    memstream_encoder<<<1, 256, 0, stream>>>(data, mean, stdv, W, bias, enc);
    memstream_dist<<<1024, 256, 0, stream>>>(memory, enc, dist);
    memstream_topk<<<1, 1024, 0, stream>>>(dist, exp_w, out);
}